// MyLayer_27462020891295
// MI455X (gfx1250) — compile-verified
//
#include <hip/hip_runtime.h>

typedef __attribute__((ext_vector_type(2))) float v2f;
typedef __attribute__((ext_vector_type(8))) float v8f;

static constexpr int T_LEN = 2048;
static constexpr int BATCH = 512;
static constexpr int EMB_D = 16;
static constexpr int HID_D = 32;
static constexpr int GATE  = 128;   // 4*HID

__device__ __forceinline__ float fsigmoid(float x) {
    // 1/(1+e^-x)
    return __fdividef(1.0f, 1.0f + __expf(-x));
}
__device__ __forceinline__ float ftanh(float x) {
    // tanh(x) = 1 - 2/(e^{2x}+1)  (exact identity; saturates correctly at +-1)
    return 1.0f - __fdividef(2.0f, __expf(2.0f * x) + 1.0f);
}

__device__ __forceinline__ v8f wmma4(v2f a, v2f b, v8f c) {
    // D = A(16x4 f32) * B(4x16 f32) + C(16x16 f32)
    return __builtin_amdgcn_wmma_f32_16x16x4_f32(
        /*neg_a=*/false, a, /*neg_b=*/false, b,
        /*c_mod=*/(short)0, c, /*reuse_a=*/false, /*reuse_b=*/false);
}

// One wave (32 lanes) owns a tile of 16 batch rows and scans all T steps,
// keeping h/c in VGPRs. Transposed weights are staged in LDS once; the
// compiler promotes the loop-invariant B-fragments into VGPRs, so the only
// per-step DS traffic is the h-tile layout bounce (D-layout -> A-layout).
// Bias enters as the C operand of the first WMMA of each accumulation chain,
// so no per-step accumulator initialization is needed.
__global__ __launch_bounds__(32)
void lstm_scan_kernel(const int* __restrict__ tokens,   // [B][T]
                      const float* __restrict__ Wemb,   // [VOCAB][16]
                      const float* __restrict__ Wx,     // [16][128]
                      const float* __restrict__ Wh,     // [32][128]
                      const float* __restrict__ bias,   // [128]
                      float* __restrict__ out)          // [B][32]
{
    __shared__ float kT[GATE * EMB_D];    // kT[n*16 + k]  = Wx[k][n]
    __shared__ float rT[GATE * HID_D];    // rT[n*32 + k]  = Wh[k][n]
    __shared__ float hbuf[16 * HID_D];    // h tile, [batch_row][hid]

    const int lane = threadIdx.x & 31;
    const int half = lane >> 4;       // 0: lanes 0-15, 1: lanes 16-31
    const int nloc = lane & 15;       // N index within fragment
    const int koff = half << 1;       // A/B fragment K offset (0 or 2)
    const int base = half << 3;       // C/D row offset (0 or 8)
    const int b0   = blockIdx.x << 4; // first batch row of this tile

    // Stage transposed weights into LDS (once per block; weights are tiny).
    for (int i = lane; i < GATE * EMB_D; i += 32) {
        const int n = i / EMB_D, k = i - n * EMB_D;
        kT[i] = Wx[k * GATE + n];
    }
    for (int i = lane; i < GATE * HID_D; i += 32) {
        const int n = i / HID_D, k = i - n * HID_D;
        rT[i] = Wh[k * GATE + n];
    }
    for (int i = lane; i < 16 * HID_D; i += 32) hbuf[i] = 0.0f;
    __syncthreads();

    // Loop-invariant bias C-fragments: column value per (nc, lane) in C/D
    // layout, identical across the 8 rows a lane holds.
    v8f bacc[8];
#pragma unroll
    for (int nc = 0; nc < 8; ++nc) {
        const float bv = bias[nc * 16 + nloc];
#pragma unroll
        for (int r = 0; r < 8; ++r) bacc[nc][r] = bv;
    }

    v8f cst[2] = {};   // c state, two 16-wide hid column groups, C/D layout
    v8f hst[2] = {};   // h state

    // Software pipeline: tokens 2 steps ahead, embedding fragments 1 ahead.
    const int* tokp = tokens + (b0 + nloc) * T_LEN;   // lane handles row nloc
    int tok0 = tokp[0];
    int tok1 = tokp[1];

    v2f ax[4];
    {
        const float* er = Wemb + (long)tok0 * EMB_D + koff;
#pragma unroll
        for (int kc = 0; kc < 4; ++kc) ax[kc] = *(const v2f*)(er + 4 * kc);
    }

    for (int t = 0; t < T_LEN; ++t) {
        const int tok2 = (t + 2 < T_LEN) ? tokp[t + 2] : 0;

        // Prefetch next step's x A-fragments (gathered embedding rows).
        v2f axn[4];
        {
            const float* er = Wemb + (long)tok1 * EMB_D + koff;
#pragma unroll
            for (int kc = 0; kc < 4; ++kc) axn[kc] = *(const v2f*)(er + 4 * kc);
        }

        // Mask bit per batch row of this tile (bit m == row m, duplicated hi half).
        const unsigned mb = __builtin_amdgcn_ballot_w32(tok0 != 0);

        // h A-fragments for this step (reads precede this step's hbuf writes;
        // DS ops are in-order within a wave, single wave owns hbuf).
        v2f ah[8];
#pragma unroll
        for (int kc = 0; kc < 8; ++kc)
            ah[kc] = *(const v2f*)&hbuf[nloc * HID_D + 4 * kc + koff];

        // All 8 gate-column accumulators live at once; 8 independent WMMA
        // accumulation chains interleave to hide XDL accumulation latency.
        // First x K-chunk seeds each chain with the bias fragment as C.
        v8f acc[8];
#pragma unroll
        for (int nc = 0; nc < 8; ++nc) {
            const float* kb = &kT[(nc * 16 + nloc) * EMB_D + koff];
            acc[nc] = wmma4(ax[0], *(const v2f*)(kb), bacc[nc]);
        }
#pragma unroll
        for (int kc = 1; kc < 4; ++kc) {     // x @ Wx, K = 16 (chunks 1..3)
#pragma unroll
            for (int nc = 0; nc < 8; ++nc) {
                const float* kb = &kT[(nc * 16 + nloc) * EMB_D + koff];
                acc[nc] = wmma4(ax[kc], *(const v2f*)(kb + 4 * kc), acc[nc]);
            }
        }
#pragma unroll
        for (int kc = 0; kc < 8; ++kc) {     // h @ Wh, K = 32
#pragma unroll
            for (int nc = 0; nc < 8; ++nc) {
                const float* rb = &rT[(nc * 16 + nloc) * HID_D + koff];
                acc[nc] = wmma4(ah[kc], *(const v2f*)(rb + 4 * kc), acc[nc]);
            }
        }

        // Gate math + masked state update (C/D layout, row = base + r).
        // Gate q covers columns [32q, 32q+32): nc = 2q + hc.
#pragma unroll
        for (int hc = 0; hc < 2; ++hc) {
#pragma unroll
            for (int r = 0; r < 8; ++r) {
                const float iv = fsigmoid(acc[0 + hc][r]);
                const float fv = fsigmoid(acc[2 + hc][r]);
                const float gv = ftanh(acc[4 + hc][r]);
                const float ov = fsigmoid(acc[6 + hc][r]);
                const float cn = fv * cst[hc][r] + iv * gv;
                const float hn = ov * ftanh(cn);
                const bool  m  = (mb >> (base + r)) & 1u;
                const float cw = m ? cn : cst[hc][r];
                const float hw = m ? hn : hst[hc][r];
                cst[hc][r] = cw;
                hst[hc][r] = hw;
                hbuf[(base + r) * HID_D + hc * 16 + nloc] = hw;
            }
        }

        tok0 = tok1;
        tok1 = tok2;
#pragma unroll
        for (int kc = 0; kc < 4; ++kc) ax[kc] = axn[kc];
    }

    // Final h (B, 32)
#pragma unroll
    for (int hc = 0; hc < 2; ++hc)
#pragma unroll
        for (int r = 0; r < 8; ++r)
            out[(b0 + base + r) * HID_D + hc * 16 + nloc] = hst[hc][r];
}

extern "C" void kernel_launch(void* const* d_in, const int* in_sizes, int n_in,
                              void* d_out, int out_size, void* d_ws, size_t ws_size,
                              hipStream_t stream) {
    const int*   tokens = (const int*)  d_in[0];
    const float* emb    = (const float*)d_in[1];
    const float* Wx     = (const float*)d_in[2];
    const float* Wh     = (const float*)d_in[3];
    const float* bias   = (const float*)d_in[4];
    float*       out    = (float*)d_out;
    (void)in_sizes; (void)n_in; (void)out_size; (void)d_ws; (void)ws_size;

    dim3 grid(BATCH / 16);   // 32 independent batch tiles, one wave each
    dim3 block(32);
    lstm_scan_kernel<<<grid, block, 0, stream>>>(tokens, emb, Wx, Wh, bias, out);
}